// ContrastiveSemLoss_86629490360546
// MI455X (gfx1250) — compile-verified
//
#include <hip/hip_runtime.h>
#include <hip/hip_bf16.h>

typedef __attribute__((ext_vector_type(2))) float v2f;
typedef __attribute__((ext_vector_type(8))) float v8f;

#define S_N     150000
#define DIM     512
#define NUSE    100
#define MROWS   112
#define NBANDS  7
#define KSTEPS  (DIM / 4)
#define TALF    0.07f
#define MAXEX   (1.0f / TALF)          // hard bound on ex (|cos| <= 1)
#define LOG2E   1.4426950408889634f

__device__ __forceinline__ v8f wmma_f32x4(v2f a, v2f b, v8f c) {
    // V_WMMA_F32_16X16X4_F32 : D = A(16x4) * B(4x16) + C(16x16), wave32
    return __builtin_amdgcn_wmma_f32_16x16x4_f32(
        /*neg_a=*/false, a, /*neg_b=*/false, b,
        /*c_mod=*/(short)0, c, /*reuse_a=*/false, /*reuse_b=*/false);
}

// ---------------------------------------------------------------------------
// Kernel 0: per-row feature norms -> rscale[n] = 1/(TAL*||fea_n||); zero accums
// grid: 112 blocks x 32 threads (one wave per row)
// ---------------------------------------------------------------------------
__global__ void csl_prep(const float* __restrict__ fea,
                         float* __restrict__ rscale,
                         float* __restrict__ rowsum,
                         float* __restrict__ pos) {
    const int row  = blockIdx.x;
    const int lane = threadIdx.x;
    if (lane == 0) { rowsum[row] = 0.f; pos[row] = 0.f; }
    if (row < NUSE) {
        float s = 0.f;
        for (int k = lane; k < DIM; k += 32) {
            float x = fea[row * DIM + k];
            s = fmaf(x, x, s);
        }
        s += __shfl_xor(s, 16, 32);
        s += __shfl_xor(s,  8, 32);
        s += __shfl_xor(s,  4, 32);
        s += __shfl_xor(s,  2, 32);
        s += __shfl_xor(s,  1, 32);
        if (lane == 0) rscale[row] = 1.0f / (TALF * sqrtf(s));
    }
}

// ---------------------------------------------------------------------------
// Kernel 1: pack A (fea, scaled) into WMMA 16x4 f32 fragment layout in ws.
//   A frag layout: lanes 0-15 -> (M=lane, K=k0+0/+1), lanes 16-31 -> (M=lane-16, K=k0+2/+3)
// grid: 112 blocks x 256 threads  (896 frags * 32 lanes)
// ---------------------------------------------------------------------------
__global__ void csl_pack(const float* __restrict__ fea,
                         const float* __restrict__ rscale,
                         float* __restrict__ Apk) {
    const int g    = blockIdx.x * 256 + threadIdx.x;
    const int lane = g & 31;
    const int frag = g >> 5;                 // 0..895
    const int ks   = frag / NBANDS;
    const int band = frag % NBANDS;
    const int row  = band * 16 + (lane & 15);
    const int k    = ks * 4 + (lane >> 4) * 2;
    v2f a;
    if (row < NUSE) {
        const float r = rscale[row];
        a.x = fea[row * DIM + k]     * r;
        a.y = fea[row * DIM + k + 1] * r;
    } else {
        a.x = 0.f; a.y = 0.f;
    }
    ((v2f*)Apk)[frag * 32 + lane] = a;
}

// ---------------------------------------------------------------------------
// Kernel 2: fused GEMM + column norms + masked exp + row LSE partials + pos pick
// grid: ceil(150000/128) blocks x 256 threads (8 waves, 16 s-columns each)
// ---------------------------------------------------------------------------
template <bool PACKED>
__global__ __launch_bounds__(256)
void csl_main(const float* __restrict__ att,
              const float* __restrict__ fea,
              const int*   __restrict__ label,
              const float* __restrict__ rscale,
              const float* __restrict__ Apk,
              float* __restrict__ rowsum,
              float* __restrict__ pos) {
    __shared__ float lds_rowsum[MROWS];
    __shared__ int   lds_label[MROWS];

    const int t = threadIdx.x;
    if (t < MROWS) {
        lds_rowsum[t] = 0.f;
        lds_label[t]  = (t < NUSE) ? label[t] : -1;
    }
    __syncthreads();

    const int  lane  = t & 31;
    const int  wave  = t >> 5;
    const int  half  = lane >> 4;            // 0 or 1
    const int  l16   = lane & 15;
    const int  col   = blockIdx.x * 128 + wave * 16 + l16;  // s index owned by lane
    const bool valid = (col < S_N);
    const int  colc  = valid ? col : (S_N - 1);             // clamp to stay in bounds
    const int  kOff  = half * 2;

    v8f acc[NBANDS];
    const v8f zero8 = {0.f, 0.f, 0.f, 0.f, 0.f, 0.f, 0.f, 0.f};
#pragma unroll
    for (int i = 0; i < NBANDS; ++i) acc[i] = zero8;

    // fallback-path row scales (A straight from global)
    float frs[NBANDS];
    int   frow[NBANDS];
    if (!PACKED) {
#pragma unroll
        for (int b = 0; b < NBANDS; ++b) {
            const int r = b * 16 + l16;
            frow[b] = r;
            frs[b]  = (r < NUSE) ? rscale[r] : 0.f;
        }
    }

    const v2f* __restrict__ Af = (const v2f*)Apk;
    float ssq = 0.f;

    for (int ks = 0; ks < KSTEPS; ++ks) {
        const int k0 = ks * 4;
        // B fragment (4x16 f32): lane owns column `col`, rows k0+kOff, k0+kOff+1
        v2f b;
        b.x = att[(k0 + kOff)     * S_N + colc];
        b.y = att[(k0 + kOff + 1) * S_N + colc];
        ssq = fmaf(b.x, b.x, ssq);              // fused ||att_col||^2 accumulation
        ssq = fmaf(b.y, b.y, ssq);
#pragma unroll
        for (int bd = 0; bd < NBANDS; ++bd) {
            v2f a;
            if (PACKED) {
                a = Af[(ks * NBANDS + bd) * 32 + lane];
            } else {
                const int kk = k0 + kOff;
                a.x = fea[frow[bd] * DIM + kk]     * frs[bd];
                a.y = fea[frow[bd] * DIM + kk + 1] * frs[bd];
            }
            acc[bd] = wmma_f32x4(a, b, acc[bd]);
        }
    }

    // full column sum-of-squares: lanes L and L+16 hold complementary K halves
    const float sumsq = ssq + __shfl_xor(ssq, 16, 32);
    const float rinv  = rsqrtf(sumsq);       // 1/||att_col||

    // Epilogue: ex = C*rinv ; e = exp(ex - 1/TAL) ; reduce per row
#pragma unroll
    for (int bd = 0; bd < NBANDS; ++bd) {
#pragma unroll
        for (int v = 0; v < 8; ++v) {
            const int   row = bd * 16 + half * 8 + v;   // C/D layout: VGPR v, halves
            const float ex  = acc[bd][v] * rinv;
            if (valid && row < NUSE && lds_label[row] == col) pos[row] = ex;
            float e = valid ? exp2f((ex - MAXEX) * LOG2E) : 0.f;
            // butterfly over the 16 lanes holding this row's columns
            e += __shfl_xor(e, 1, 32);
            e += __shfl_xor(e, 2, 32);
            e += __shfl_xor(e, 4, 32);
            e += __shfl_xor(e, 8, 32);
            if (l16 == 0 && row < NUSE) atomicAdd(&lds_rowsum[row], e);
        }
    }
    __syncthreads();
    if (t < NUSE) atomicAdd(&rowsum[t], lds_rowsum[t]);
}

// ---------------------------------------------------------------------------
// Kernel 3: loss = (1/100) * sum_n ( log(rowsum[n]) + 1/TAL - pos[n] )
// ---------------------------------------------------------------------------
__global__ void csl_finalize(const float* __restrict__ rowsum,
                             const float* __restrict__ pos,
                             float* __restrict__ out) {
    __shared__ float red[128];
    const int t = threadIdx.x;
    float v = 0.f;
    if (t < NUSE) v = logf(rowsum[t]) + MAXEX - pos[t];
    red[t] = v;
    __syncthreads();
    for (int s = 64; s > 0; s >>= 1) {
        if (t < s) red[t] += red[t + s];
        __syncthreads();
    }
    if (t == 0) out[0] = red[0] / (float)NUSE;
}

// ---------------------------------------------------------------------------
extern "C" void kernel_launch(void* const* d_in, const int* in_sizes, int n_in,
                              void* d_out, int out_size, void* d_ws, size_t ws_size,
                              hipStream_t stream) {
    const float* att   = (const float*)d_in[0];   // (512, 150000)
    const float* fea   = (const float*)d_in[1];   // (4096, 512)
    const int*   label = (const int*)d_in[2];     // (4096,)
    float*       out   = (float*)d_out;

    float* ws     = (float*)d_ws;
    float* rscale = ws;          // 112 floats
    float* rowsum = ws + 112;    // 112 floats
    float* pos    = ws + 224;    // 112 floats
    float* Apk    = ws + 384;    // 128*7*64 = 57344 floats (229 KB), 16B-aligned offset

    const size_t need  = (size_t)(384 + KSTEPS * NBANDS * 64) * sizeof(float);
    const bool packed  = (ws_size >= need);
    const int  nTiles  = (S_N + 127) / 128;   // 1172

    csl_prep<<<MROWS, 32, 0, stream>>>(fea, rscale, rowsum, pos);
    if (packed) {
        csl_pack<<<112, 256, 0, stream>>>(fea, rscale, Apk);
        csl_main<true><<<nTiles, 256, 0, stream>>>(att, fea, label, rscale, Apk, rowsum, pos);
    } else {
        csl_main<false><<<nTiles, 256, 0, stream>>>(att, fea, label, rscale, Apk, rowsum, pos);
    }
    csl_finalize<<<1, 128, 0, stream>>>(rowsum, pos, out);
}